// GNN_79465484910998
// MI455X (gfx1250) — compile-verified
//
#include <hip/hip_runtime.h>
#include <hip/hip_bf16.h>

typedef __attribute__((ext_vector_type(2))) float v2f;
typedef __attribute__((ext_vector_type(8))) float v8f;

#define GNN_N    100000
#define GNN_E    1600000
#define GNN_NB   5000
#define GNN_D    64
#define GNN_NBLK 512

// ---------- float <-> ordered-uint encoding for atomic min/max on floats ----
__device__ __forceinline__ unsigned fenc(float f) {
    unsigned u = __float_as_uint(f);
    return (u & 0x80000000u) ? ~u : (u | 0x80000000u);
}
__device__ __forceinline__ float fdec(unsigned k) {
    return (k & 0x80000000u) ? __uint_as_float(k ^ 0x80000000u)
                             : __uint_as_float(~k);
}
__device__ __forceinline__ float lrelu(float x) { return x > 0.0f ? x : 0.01f * x; }
__device__ __forceinline__ float silu(float x)  { return x / (1.0f + __expf(-x)); }

// ---------- 0. boundary displacement normalization (scatter, same-value dup ok)
__global__ void k_boundary(float* __restrict__ xin, const float* __restrict__ nodes,
                           const int* __restrict__ bnodes,
                           const float* __restrict__ mean, const float* __restrict__ stdv,
                           int nb) {
    int i = blockIdx.x * blockDim.x + threadIdx.x;
    if (i >= nb) return;
    int n = bnodes[i];
#pragma unroll
    for (int j = 0; j < 3; ++j)
        xin[n * 7 + 3 + j] = (nodes[n * 7 + 3 + j] - mean[j]) / stdv[j];
}

// ---------- 1. embedding: x = xin(N,7) @ eW(7,64) + eb ----------------------
__global__ void k_embed(const float* __restrict__ xin, const float* __restrict__ eW,
                        const float* __restrict__ eb, float* __restrict__ x, int M) {
    int idx = blockIdx.x * blockDim.x + threadIdx.x;
    if (idx >= M * GNN_D) return;
    int n = idx >> 6, d = idx & 63;
    float acc = eb[d];
#pragma unroll
    for (int k = 0; k < 7; ++k) acc += xin[n * 7 + k] * eW[k * GNN_D + d];
    x[idx] = acc;
}

// ---------- 2. h = x(M,64) @ W(64,64)  via V_WMMA_F32_16X16X4_F32 -----------
// one wave (32 lanes) per 16x16 output tile, K swept in steps of 4 (16 WMMAs)
__global__ void k_gemm64_wmma(const float* __restrict__ x, const float* __restrict__ W,
                              float* __restrict__ h) {
    const int mtile = blockIdx.x * 16;
    const int ntile = blockIdx.y * 16;
    const int lane = threadIdx.x;      // 0..31, blockDim.x == 32, EXEC all ones
    const int l16 = lane & 15;
    const int hi  = lane >> 4;         // 0: K pair {k,k+1}, 1: K pair {k+2,k+3}

    v8f acc = {};
#pragma unroll
    for (int k = 0; k < GNN_D; k += 4) {
        const int kb = k + hi * 2;
        // A fragment: row mtile+l16, K = kb, kb+1 (contiguous in row-major x)
        v2f a;
        const float* ap = x + (size_t)(mtile + l16) * GNN_D + kb;
        a.x = ap[0]; a.y = ap[1];
        // B fragment: col ntile+l16, K = kb, kb+1 (row-major W, stride 64)
        v2f b;
        b.x = W[(size_t)kb * GNN_D + ntile + l16];
        b.y = W[(size_t)(kb + 1) * GNN_D + ntile + l16];
        acc = __builtin_amdgcn_wmma_f32_16x16x4_f32(false, a, false, b,
                                                    (short)0, acc, false, false);
    }
    // C/D layout: VGPR i -> (M = i + 8*hi, N = l16)
#pragma unroll
    for (int i = 0; i < 8; ++i)
        h[(size_t)(mtile + hi * 8 + i) * GNN_D + ntile + l16] = acc[i];
}

// ---------- 3. per-node attention dots: hsA = h.A[:64], hrA = h.A[64:] ------
__global__ void k_node_dots(const float* __restrict__ h, const float* __restrict__ A,
                            float* __restrict__ hsA, float* __restrict__ hrA, int M) {
    int n = blockIdx.x * blockDim.x + threadIdx.x;
    if (n >= M) return;
    float sa = 0.0f, sb = 0.0f;
    const float* hp = h + (size_t)n * GNN_D;
#pragma unroll
    for (int d = 0; d < GNN_D; ++d) {
        float v = hp[d];
        sa += v * A[d];
        sb += v * A[GNN_D + d];
    }
    hsA[n] = sa; hrA[n] = sb;
}

// ---------- 4. edge pass 1: segment max of leaky_relu scores over receivers -
__global__ void k_edge_max(const float* __restrict__ hsA, const float* __restrict__ hrA,
                           const int* __restrict__ s, const int* __restrict__ r,
                           const int* __restrict__ bmap, unsigned* __restrict__ mkey,
                           int E) {
    int e = blockIdx.x * blockDim.x + threadIdx.x;
    if (e >= E) return;
    int ss = s[e], rr = r[e];
    if (bmap) { ss = bmap[ss]; rr = bmap[rr]; }
    float sc = lrelu(hsA[ss] + hrA[rr]);
    atomicMax(&mkey[rr], fenc(sc));
}

// ---------- 5. edge pass 2: wave per edge, accumulate exp-weighted features -
__global__ void k_edge_accum(const float* __restrict__ h,
                             const float* __restrict__ hsA, const float* __restrict__ hrA,
                             const int* __restrict__ s, const int* __restrict__ r,
                             const int* __restrict__ bmap,
                             const unsigned* __restrict__ mkey,
                             float* __restrict__ ssum, float* __restrict__ out, int E) {
    int wid  = threadIdx.x >> 5;                  // wave in block (blockDim=256 -> 8)
    int lane = threadIdx.x & 31;
    int e = blockIdx.x * 8 + wid;
    if (e >= E) return;
    int ss = s[e], rr = r[e];
    if (bmap) { ss = bmap[ss]; rr = bmap[rr]; }
    float sc = lrelu(hsA[ss] + hrA[rr]);
    float w = __expf(sc - fdec(mkey[rr]));
    if (lane == 0) atomicAdd(&ssum[rr], w);
    const float* hp = h + (size_t)ss * GNN_D;
    float* op = out + (size_t)rr * GNN_D;
    atomicAdd(&op[lane],      hp[lane]      * w);
    atomicAdd(&op[lane + 32], hp[lane + 32] * w);
}

// ---------- 6. divide by softmax denominator, optional residual add ---------
__global__ void k_div(float* __restrict__ out, const float* __restrict__ ssum,
                      const float* __restrict__ res, int M) {
    int idx = blockIdx.x * blockDim.x + threadIdx.x;
    if (idx >= M * GNN_D) return;
    int n = idx >> 6;
    float den = ssum[n];
    float v = den > 0.0f ? out[idx] / den : 0.0f;
    if (res) v += res[idx];
    out[idx] = v;
}

// ---------- 7. column statistics for graphnorm ------------------------------
__global__ void k_colstats(const float* __restrict__ x, float* __restrict__ csum,
                           float* __restrict__ csq, int M) {
    __shared__ float shs[4][GNN_D];
    __shared__ float shq[4][GNN_D];
    int d = threadIdx.x & 63;
    int g = threadIdx.x >> 6;                     // 0..3
    int base = blockIdx.x * 64;
    float s = 0.0f, q = 0.0f;
    for (int i = g; i < 64; i += 4) {
        int n = base + i;
        if (n < M) {
            float v = x[(size_t)n * GNN_D + d];
            s += v; q += v * v;
        }
    }
    shs[g][d] = s; shq[g][d] = q;
    __syncthreads();
    if (g == 0) {
        s = shs[0][d] + shs[1][d] + shs[2][d] + shs[3][d];
        q = shq[0][d] + shq[1][d] + shq[2][d] + shq[3][d];
        atomicAdd(&csum[d], s);
        atomicAdd(&csq[d], q);
    }
}

// ---------- 8. graphnorm + silu + residual ----------------------------------
__global__ void k_norm_silu_res(float* __restrict__ x, const float* __restrict__ res,
                                const float* __restrict__ gamma, const float* __restrict__ beta,
                                const float* __restrict__ csum, const float* __restrict__ csq,
                                int M) {
    int idx = blockIdx.x * blockDim.x + threadIdx.x;
    if (idx >= M * GNN_D) return;
    int d = idx & 63;
    float mean = csum[d] / (float)M;
    float var  = csq[d] / (float)M - mean * mean;
    float y = gamma[d] * (x[idx] - mean) * rsqrtf(var + 1e-5f) + beta[d];
    x[idx] = silu(y) + res[idx];
}

// ---------- 9. coordinate min/max -------------------------------------------
__global__ void k_minmax(const float* __restrict__ nodes, unsigned* __restrict__ mm, int M) {
    int n = blockIdx.x * blockDim.x + threadIdx.x;
    if (n >= M) return;
#pragma unroll
    for (int j = 0; j < 3; ++j) {
        unsigned k = fenc(nodes[n * 7 + j]);
        atomicMin(&mm[j], k);        // mn keys
        atomicMax(&mm[3 + j], k);    // mx keys
    }
}

// ---------- 10. block id per node -------------------------------------------
__global__ void k_bid(const float* __restrict__ nodes, const unsigned* __restrict__ mm,
                      int* __restrict__ bid, int M) {
    int n = blockIdx.x * blockDim.x + threadIdx.x;
    if (n >= M) return;
    int gi[3];
#pragma unroll
    for (int j = 0; j < 3; ++j) {
        float mn = fdec(mm[j]), mx = fdec(mm[3 + j]);
        float cell = (mx - mn) * (1.0f / 8.0f);
        int g = 0;
        if (cell > 0.0f) {
            g = (int)floorf((nodes[n * 7 + j] - mn) / cell);
            g = g < 0 ? 0 : (g > 7 ? 7 : g);
        }
        gi[j] = g;
    }
    bid[n] = gi[0] * 64 + gi[1] * 8 + gi[2];
}

// ---------- 11. coarsen: segment-sum node features into blocks --------------
__global__ void k_coarsen(const float* __restrict__ x, const int* __restrict__ bid,
                          float* __restrict__ cnodes, float* __restrict__ sizes, int M) {
    int wid  = threadIdx.x >> 5;
    int lane = threadIdx.x & 31;
    int n = blockIdx.x * 8 + wid;
    if (n >= M) return;
    int b = bid[n];
    const float* xp = x + (size_t)n * GNN_D;
    float* cp = cnodes + (size_t)b * GNN_D;
    atomicAdd(&cp[lane],      xp[lane]);
    atomicAdd(&cp[lane + 32], xp[lane + 32]);
    if (lane == 0) atomicAdd(&sizes[b], 1.0f);
}

__global__ void k_coarse_scale(float* __restrict__ cnodes, const float* __restrict__ sizes) {
    int idx = blockIdx.x * blockDim.x + threadIdx.x;
    if (idx >= GNN_NBLK * GNN_D) return;
    int b = idx >> 6;
    cnodes[idx] = cnodes[idx] * rsqrtf(sizes[b] + 1e-10f);
}

// ---------- 12. final reduction & decode ------------------------------------
__global__ void k_final(const float* __restrict__ x2, const float* __restrict__ decW,
                        const float* __restrict__ decb, float* __restrict__ out) {
    __shared__ float sh[GNN_D];
    int d = threadIdx.x;                           // blockDim.x == 64
    float s = 0.0f;
    for (int n = 0; n < GNN_NBLK; ++n) s += x2[(size_t)n * GNN_D + d];
    sh[d] = s * decW[d];
    __syncthreads();
    for (int off = 32; off > 0; off >>= 1) {
        if (d < off) sh[d] += sh[d + off];
        __syncthreads();
    }
    if (d == 0) out[0] = sh[0] + decb[0];
}

// ============================================================================
extern "C" void kernel_launch(void* const* d_in, const int* in_sizes, int n_in,
                              void* d_out, int out_size, void* d_ws, size_t ws_size,
                              hipStream_t stream) {
    const float* nodes  = (const float*)d_in[0];
    const int*   send   = (const int*)d_in[1];
    const int*   recv   = (const int*)d_in[2];
    const int*   bnodes = (const int*)d_in[3];
    const float* dmean  = (const float*)d_in[4];
    const float* dstd   = (const float*)d_in[5];
    const float* eW     = (const float*)d_in[6];
    const float* eb     = (const float*)d_in[7];
    const float* W1 = (const float*)d_in[8];  const float* A1 = (const float*)d_in[9];
    const float* g1 = (const float*)d_in[10]; const float* b1 = (const float*)d_in[11];
    const float* W2 = (const float*)d_in[12]; const float* A2 = (const float*)d_in[13];
    const float* g2 = (const float*)d_in[14]; const float* b2 = (const float*)d_in[15];
    const float* W3 = (const float*)d_in[16]; const float* A3 = (const float*)d_in[17];
    const float* dW = (const float*)d_in[18]; const float* db = (const float*)d_in[19];

    // ---- workspace carve-up (256B aligned) ----
    char* w = (char*)d_ws;
    size_t off = 0;
    auto carve = [&](size_t bytes) -> char* {
        char* p = w + off;
        off = (off + bytes + 255) & ~(size_t)255;
        return p;
    };
    float*    xin    = (float*)   carve((size_t)GNN_N * 7 * 4);
    float*    x      = (float*)   carve((size_t)GNN_N * GNN_D * 4);
    float*    res    = (float*)   carve((size_t)GNN_N * GNN_D * 4);
    float*    h      = (float*)   carve((size_t)GNN_N * GNN_D * 4);
    float*    hsA    = (float*)   carve((size_t)GNN_N * 4);
    float*    hrA    = (float*)   carve((size_t)GNN_N * 4);
    unsigned* mkey   = (unsigned*)carve((size_t)GNN_N * 4);
    float*    ssum   = (float*)   carve((size_t)GNN_N * 4);
    int*      bid    = (int*)     carve((size_t)GNN_N * 4);
    float*    csum   = (float*)   carve(GNN_D * 4);
    float*    csq    = (float*)   carve(GNN_D * 4);
    unsigned* mm     = (unsigned*)carve(6 * 4);
    float*    x2     = (float*)   carve((size_t)GNN_NBLK * GNN_D * 4);  // cnodes
    float*    sizes  = (float*)   carve((size_t)GNN_NBLK * 4);
    float*    res2   = (float*)   carve((size_t)GNN_NBLK * GNN_D * 4);
    float*    h2     = (float*)   carve((size_t)GNN_NBLK * GNN_D * 4);
    (void)ws_size; (void)n_in; (void)in_sizes; (void)out_size;

    const int T = 256;
    const int ND = GNN_N * GNN_D;
    const int CD = GNN_NBLK * GNN_D;

    // ---- input features + boundary normalization ----
    hipMemcpyAsync(xin, nodes, (size_t)GNN_N * 7 * 4, hipMemcpyDeviceToDevice, stream);
    k_boundary<<<(GNN_NB + T - 1) / T, T, 0, stream>>>(xin, nodes, bnodes, dmean, dstd, GNN_NB);

    // ---- embedding ----
    k_embed<<<(ND + T - 1) / T, T, 0, stream>>>(xin, eW, eb, x, GNN_N);

    // ---- GAT layer 1 (fine graph, identity map) ----
    hipMemcpyAsync(res, x, (size_t)ND * 4, hipMemcpyDeviceToDevice, stream);
    k_gemm64_wmma<<<dim3(GNN_N / 16, 4), 32, 0, stream>>>(x, W1, h);
    k_node_dots<<<(GNN_N + T - 1) / T, T, 0, stream>>>(h, A1, hsA, hrA, GNN_N);
    hipMemsetAsync(mkey, 0, (size_t)GNN_N * 4, stream);
    hipMemsetAsync(ssum, 0, (size_t)GNN_N * 4, stream);
    hipMemsetAsync(x, 0, (size_t)ND * 4, stream);
    k_edge_max<<<(GNN_E + T - 1) / T, T, 0, stream>>>(hsA, hrA, send, recv, nullptr, mkey, GNN_E);
    k_edge_accum<<<(GNN_E + 7) / 8, T, 0, stream>>>(h, hsA, hrA, send, recv, nullptr, mkey, ssum, x, GNN_E);
    k_div<<<(ND + T - 1) / T, T, 0, stream>>>(x, ssum, nullptr, GNN_N);
    hipMemsetAsync(csum, 0, GNN_D * 4, stream);
    hipMemsetAsync(csq, 0, GNN_D * 4, stream);
    k_colstats<<<(GNN_N + 63) / 64, T, 0, stream>>>(x, csum, csq, GNN_N);
    k_norm_silu_res<<<(ND + T - 1) / T, T, 0, stream>>>(x, res, g1, b1, csum, csq, GNN_N);

    // ---- coarsening ----
    hipMemsetAsync(mm, 0xFF, 3 * 4, stream);       // mn keys -> UINT_MAX
    hipMemsetAsync(mm + 3, 0x00, 3 * 4, stream);   // mx keys -> 0
    k_minmax<<<(GNN_N + T - 1) / T, T, 0, stream>>>(nodes, mm, GNN_N);
    k_bid<<<(GNN_N + T - 1) / T, T, 0, stream>>>(nodes, mm, bid, GNN_N);
    hipMemsetAsync(x2, 0, (size_t)CD * 4, stream);
    hipMemsetAsync(sizes, 0, (size_t)GNN_NBLK * 4, stream);
    k_coarsen<<<(GNN_N + 7) / 8, T, 0, stream>>>(x, bid, x2, sizes, GNN_N);
    k_coarse_scale<<<(CD + T - 1) / T, T, 0, stream>>>(x2, sizes);

    // ---- GAT layer 2 (coarse graph, bid map) ----
    hipMemcpyAsync(res2, x2, (size_t)CD * 4, hipMemcpyDeviceToDevice, stream);
    k_gemm64_wmma<<<dim3(GNN_NBLK / 16, 4), 32, 0, stream>>>(x2, W2, h2);
    k_node_dots<<<(GNN_NBLK + T - 1) / T, T, 0, stream>>>(h2, A2, hsA, hrA, GNN_NBLK);
    hipMemsetAsync(mkey, 0, (size_t)GNN_NBLK * 4, stream);
    hipMemsetAsync(ssum, 0, (size_t)GNN_NBLK * 4, stream);
    hipMemsetAsync(x2, 0, (size_t)CD * 4, stream);
    k_edge_max<<<(GNN_E + T - 1) / T, T, 0, stream>>>(hsA, hrA, send, recv, bid, mkey, GNN_E);
    k_edge_accum<<<(GNN_E + 7) / 8, T, 0, stream>>>(h2, hsA, hrA, send, recv, bid, mkey, ssum, x2, GNN_E);
    k_div<<<(CD + T - 1) / T, T, 0, stream>>>(x2, ssum, nullptr, GNN_NBLK);
    hipMemsetAsync(csum, 0, GNN_D * 4, stream);
    hipMemsetAsync(csq, 0, GNN_D * 4, stream);
    k_colstats<<<(GNN_NBLK + 63) / 64, T, 0, stream>>>(x2, csum, csq, GNN_NBLK);
    k_norm_silu_res<<<(CD + T - 1) / T, T, 0, stream>>>(x2, res2, g2, b2, csum, csq, GNN_NBLK);

    // ---- GAT layer 3 (coarse graph, no norm, residual only) ----
    hipMemcpyAsync(res2, x2, (size_t)CD * 4, hipMemcpyDeviceToDevice, stream);
    k_gemm64_wmma<<<dim3(GNN_NBLK / 16, 4), 32, 0, stream>>>(x2, W3, h2);
    k_node_dots<<<(GNN_NBLK + T - 1) / T, T, 0, stream>>>(h2, A3, hsA, hrA, GNN_NBLK);
    hipMemsetAsync(mkey, 0, (size_t)GNN_NBLK * 4, stream);
    hipMemsetAsync(ssum, 0, (size_t)GNN_NBLK * 4, stream);
    hipMemsetAsync(x2, 0, (size_t)CD * 4, stream);
    k_edge_max<<<(GNN_E + T - 1) / T, T, 0, stream>>>(hsA, hrA, send, recv, bid, mkey, GNN_E);
    k_edge_accum<<<(GNN_E + 7) / 8, T, 0, stream>>>(h2, hsA, hrA, send, recv, bid, mkey, ssum, x2, GNN_E);
    k_div<<<(CD + T - 1) / T, T, 0, stream>>>(x2, ssum, res2, GNN_NBLK);

    // ---- aggregate + decode ----
    k_final<<<1, GNN_D, 0, stream>>>(x2, dW, db, (float*)d_out);
}